// Entmax_4071628997037
// MI455X (gfx1250) — compile-verified
//
#include <hip/hip_runtime.h>
#include <math.h>

#ifndef __has_builtin
#define __has_builtin(x) 0
#endif

#define COLS    32000
#define THREADS 1024
#define VPT     8                     // float4 vectors per thread (32 floats)
#define PADN    (THREADS * 4 * VPT)   // 32768 floats staged per row
#define NITER   30                    // bisection iterations (interval len 1 -> ~1e-9)

#if __has_builtin(__builtin_amdgcn_global_load_async_to_lds_b128)
#define HAVE_ASYNC_LDS 1
#else
#define HAVE_ASYNC_LDS 0
#endif

typedef int v4i __attribute__((__vector_size__(4 * sizeof(int))));
typedef __attribute__((address_space(1))) v4i* gv4i_ptr;   // global int4*
typedef __attribute__((address_space(3))) v4i* lv4i_ptr;   // LDS int4*

__global__ __launch_bounds__(THREADS)
void entmax15_kernel(const float* __restrict__ x, float* __restrict__ y) {
    __shared__ float sdata[PADN];     // 128 KB row staging
    __shared__ float wmax[32];
    __shared__ float wsum[2][32];     // double-buffered cross-wave sums

    const int tid  = threadIdx.x;
    const int lane = tid & 31;
    const int wv   = tid >> 5;
    const long long row = blockIdx.x;

    const float* xrow = x + row * (long long)COLS;
    float*       yrow = y + row * (long long)COLS;

    // Pad tail [COLS, PADN) with -inf so it contributes nothing.
    for (int i = COLS + tid; i < PADN; i += THREADS)
        sdata[i] = -INFINITY;

    // Stage the row into LDS via async global->LDS b128 (CDNA5 path).
#pragma unroll
    for (int j = 0; j < VPT; ++j) {
        const int e = j * (THREADS * 4) + tid * 4;
        if (e < COLS) {
#if HAVE_ASYNC_LDS
            __builtin_amdgcn_global_load_async_to_lds_b128(
                (gv4i_ptr)(xrow + e),
                (lv4i_ptr)(sdata + e),
                0, 0);
#else
            *(float4*)(sdata + e) = *(const float4*)(xrow + e);
#endif
        }
    }
#if HAVE_ASYNC_LDS
#if __has_builtin(__builtin_amdgcn_s_wait_asynccnt)
    __builtin_amdgcn_s_wait_asynccnt(0);
#else
    asm volatile("s_wait_asynccnt 0" ::: "memory");
#endif
#endif
    __syncthreads();

    // LDS -> registers, tracking the per-thread max.
    float4 v[VPT];
    float m = -INFINITY;
#pragma unroll
    for (int j = 0; j < VPT; ++j) {
        v[j] = *(const float4*)(sdata + j * (THREADS * 4) + tid * 4);
        m = fmaxf(m, fmaxf(fmaxf(v[j].x, v[j].y), fmaxf(v[j].z, v[j].w)));
    }
    // wave32 max reduce
#pragma unroll
    for (int off = 16; off; off >>= 1)
        m = fmaxf(m, __shfl_xor(m, off, 32));
    if (lane == 0) wmax[wv] = m;
    __syncthreads();
    float rm = wmax[0];
#pragma unroll
    for (int w = 1; w < 32; ++w) rm = fmaxf(rm, wmax[w]);

    // s = (x - max) / 2  => max(s) == 0, tau* in [-1, 0]
#pragma unroll
    for (int j = 0; j < VPT; ++j) {
        v[j].x = (v[j].x - rm) * 0.5f;
        v[j].y = (v[j].y - rm) * 0.5f;
        v[j].z = (v[j].z - rm) * 0.5f;
        v[j].w = (v[j].w - rm) * 0.5f;
    }

    // Bisection on f(tau) = sum relu(s - tau)^2, root at f == 1.
    float lo = -1.0f, hi = 0.0f;
#pragma unroll 1
    for (int it = 0; it < NITER; ++it) {
        const float tau = 0.5f * (lo + hi);
        float s = 0.0f;
#pragma unroll
        for (int j = 0; j < VPT; ++j) {
            float r;
            r = fmaxf(v[j].x - tau, 0.0f); s = fmaf(r, r, s);
            r = fmaxf(v[j].y - tau, 0.0f); s = fmaf(r, r, s);
            r = fmaxf(v[j].z - tau, 0.0f); s = fmaf(r, r, s);
            r = fmaxf(v[j].w - tau, 0.0f); s = fmaf(r, r, s);
        }
#pragma unroll
        for (int off = 16; off; off >>= 1)
            s += __shfl_xor(s, off, 32);
        const int p = it & 1;
        if (lane == 0) wsum[p][wv] = s;
        __syncthreads();
        float tot = 0.0f;
#pragma unroll
        for (int w = 0; w < 32; ++w) tot += wsum[p][w];   // broadcast reads
        if (tot >= 1.0f) lo = tau; else hi = tau;          // uniform branch
    }
    const float tau = 0.5f * (lo + hi);

    // y = relu(s - tau)^2, b128 stores.
#pragma unroll
    for (int j = 0; j < VPT; ++j) {
        const int e = j * (THREADS * 4) + tid * 4;
        if (e < COLS) {
            float4 o; float r;
            r = fmaxf(v[j].x - tau, 0.0f); o.x = r * r;
            r = fmaxf(v[j].y - tau, 0.0f); o.y = r * r;
            r = fmaxf(v[j].z - tau, 0.0f); o.z = r * r;
            r = fmaxf(v[j].w - tau, 0.0f); o.w = r * r;
            *(float4*)(yrow + e) = o;
        }
    }
}

extern "C" void kernel_launch(void* const* d_in, const int* in_sizes, int n_in,
                              void* d_out, int out_size, void* d_ws, size_t ws_size,
                              hipStream_t stream) {
    (void)n_in; (void)d_ws; (void)ws_size; (void)out_size;
    const float* x = (const float*)d_in[0];
    float* y = (float*)d_out;
    const int rows = in_sizes[0] / COLS;   // 4096 x 32000
    entmax15_kernel<<<rows, THREADS, 0, stream>>>(x, y);
}